// SalientPixelsBCELoss_71322226917957
// MI455X (gfx1250) — compile-verified
//
#include <hip/hip_runtime.h>
#include <hip/hip_bf16.h>

// ---- problem constants (fixed by the reference generator) ----
#define BB   32                    // batch
#define NN   (512 * 512)           // tokens per batch
#define TPB  256                   // 8 waves (wave32)
#define BPB  64                    // streaming blocks per batch
#define EPT  (NN / BPB / TPB)      // 16 elements per thread
#define CHUNK (NN / BPB)           // 4096 elements per block
#define NBLK (BB * BPB)            // 2048 streaming blocks
#define COARSE_BINS 16384          // score bits [29:16] (scores in [0,1) -> bin <= 16255)
#define FINE_BINS   65536          // score bits [15:0]
#define TIE_CAP     256

typedef __attribute__((ext_vector_type(2))) float v2f;
typedef __attribute__((ext_vector_type(8))) float v8f;

// Reduce 64 values (2 per lane) across a wave32 on the matrix pipe:
// D = A(16x4) * ones(4x16) -> D[m][n] = rowsum(m). Per-lane sum of the 8 D
// VGPRs gives rows 0..7 (lanes 0-15) / rows 8..15 (lanes 16-31); one
// xor-16 shuffle finishes. Result = sum of all 64 A entries, in every lane.
__device__ __forceinline__ float wave_sum64_wmma(float a0, float a1) {
  v2f A;  A[0] = a0;   A[1] = a1;
  v2f Bm; Bm[0] = 1.f; Bm[1] = 1.f;
  v8f C = {};
  v8f D = __builtin_amdgcn_wmma_f32_16x16x4_f32(false, A, false, Bm,
                                                (short)0, C, false, false);
  float s = ((D[0] + D[1]) + (D[2] + D[3])) + ((D[4] + D[5]) + (D[6] + D[7]));
  s += __shfl_xor(s, 16, 32);
  return s;
}

// Deterministic block reduction (wmma per wave, fixed-order across waves).
// Valid on threadIdx.x == 0. Requires all 256 threads converged (EXEC all-1s).
__device__ __forceinline__ float block_sum(float a0, float a1, float* lred) {
  float w = wave_sum64_wmma(a0, a1);
  const int t = threadIdx.x;
  if ((t & 31) == 0) lred[t >> 5] = w;
  __syncthreads();
  float tot = 0.f;
  if (t == 0)
    for (int i = 0; i < TPB / 32; ++i) tot += lred[i];
  return tot;
}

// ---- CDNA5 async global->LDS staging (ASYNCcnt-tracked, §15.18.3 op 98) ----
// GVS mode: mem = SADDR(64b uniform) + VADDR(32b per-lane byte offset);
// VDST holds the LDS byte address. 16 bytes per lane per issue.
__device__ __forceinline__ void async_copy16(unsigned lds_byte_addr,
                                             const void* base, unsigned byte_off) {
  asm volatile("global_load_async_to_lds_b128 %0, %1, %2"
               :: "v"(lds_byte_addr), "v"(byte_off), "s"(base) : "memory");
}
__device__ __forceinline__ void wait_async0() {
  asm volatile("s_wait_asynccnt 0x0" ::: "memory");
}
// LDS byte offset of a __shared__ object: addrspacecast AS3->flat keeps the
// LDS offset in the low 32 bits.
__device__ __forceinline__ unsigned lds_off(const void* p) {
  return (unsigned)(unsigned long long)p;
}

__global__ void k_zero(unsigned* __restrict__ p, long n) {
  long i = (long)blockIdx.x * blockDim.x + threadIdx.x;
  if (i < n) p[i] = 0u;
}

// Pass A: fused elementwise math + Sum(logp1c) partials + delta store +
// per-batch coarse histogram of score bit patterns.
__global__ void __launch_bounds__(TPB)
k_main(const float* __restrict__ ds, const float* __restrict__ smap,
       const float* __restrict__ U, float* __restrict__ delta,
       unsigned* __restrict__ coarse, float* __restrict__ pA) {
  __shared__ unsigned lhist[COARSE_BINS / 2];  // two u16 counters per word
  __shared__ float lred[TPB / 32];
  const int t  = threadIdx.x;
  const int b  = blockIdx.x / BPB;
  const int cb = blockIdx.x % BPB;
  for (int i = t; i < COARSE_BINS / 2; i += TPB) lhist[i] = 0u;
  __syncthreads();

  const long base = (long)b * NN + (long)cb * CHUNK;
  const float2* ds2 = (const float2*)ds;
  const float2* u2  = (const float2*)U;
  float acc0 = 0.f, acc1 = 0.f;
  for (int k = 0; k < EPT; ++k) {
    const long e = base + (long)k * TPB + t;
    __builtin_prefetch(&ds2[e + TPB], 0, 0);   // global_prefetch_b8
    __builtin_prefetch(&u2[e + TPB], 0, 0);
    float2 d  = ds2[e];
    float2 uu = u2[e];
    float  s  = smap[e];
    // gumbel = -log(-log(u)); a = z0 - z1
    float g0 = -__logf(-__logf(uu.x));
    float g1 = -__logf(-__logf(uu.y));
    float a  = (d.x + g0) - (d.y + g1);
    // stable 2-class log-softmax: logp1 = -log1p(exp(a)), logp0 = a + logp1
    float lp1 = (a > 0.f) ? (-a - log1pf(__expf(-a))) : (-log1pf(__expf(a)));
    float l1c = fmaxf(lp1,     -100.f);        // torch BCE clamp
    float l0c = fmaxf(lp1 + a, -100.f);
    delta[e] = l0c - l1c;
    if (k & 1) acc1 += l1c; else acc0 += l1c;
    // ordering of non-negative floats == ordering of their bit patterns
    unsigned key = __float_as_uint(s);
    unsigned bin = key >> 16;
    if (bin >= COARSE_BINS) bin = COARSE_BINS - 1;
    atomicAdd(&lhist[bin >> 1], 1u << ((bin & 1) << 4));  // <=4096/block: no carry
  }
  __syncthreads();

  unsigned* gh = coarse + (long)b * COARSE_BINS;
  for (int i = t; i < COARSE_BINS / 2; i += TPB) {
    unsigned v = lhist[i], lo = v & 0xFFFFu, hi = v >> 16;
    if (lo) atomicAdd(&gh[2 * i], lo);
    if (hi) atomicAdd(&gh[2 * i + 1], hi);
  }
  float tot = block_sum(acc0, acc1, lred);
  if (t == 0) pA[blockIdx.x] = tot;
}

// Pass B: per-batch coarse threshold bucket + strictly-greater count G0.
__global__ void __launch_bounds__(TPB)
k_coarse_select(const unsigned* __restrict__ coarse, const int* __restrict__ Kp,
                unsigned* __restrict__ Tbucket, unsigned* __restrict__ G0) {
  __shared__ unsigned csum[TPB];
  const int b = blockIdx.x, t = threadIdx.x;
  const unsigned* h = coarse + (long)b * COARSE_BINS;
  const int per = COARSE_BINS / TPB;  // 64
  unsigned s = 0;
  for (int i = 0; i < per; ++i) s += h[t * per + i];
  csum[t] = s;
  __syncthreads();
  if (t == 0) {
    const unsigned K = (unsigned)Kp[0];
    unsigned acc = 0; int tc = 0;
    for (int c = TPB - 1; c >= 0; --c) {
      if (acc + csum[c] >= K) { tc = c; break; }
      acc += csum[c];
    }
    unsigned tb = 0;
    for (int bin = (tc + 1) * per - 1; bin >= tc * per; --bin) {
      unsigned v = h[bin];
      if (acc + v >= K) { tb = (unsigned)bin; break; }
      acc += v;
    }
    Tbucket[b] = tb;
    G0[b] = acc;
  }
}

// Pass B2a: fine histogram (low 16 bits) of elements inside threshold bucket.
// s_map chunk is staged into LDS with async b128 loads (deep prefetch, no
// VGPR pressure), then read back from LDS.
__global__ void __launch_bounds__(TPB)
k_fine_hist(const float* __restrict__ smap, const unsigned* __restrict__ Tbucket,
            unsigned* __restrict__ fine) {
  __shared__ float sbuf[CHUNK];                 // 16 KB
  const int t  = threadIdx.x;
  const int b  = blockIdx.x / BPB;
  const int cb = blockIdx.x % BPB;
  const unsigned tb = Tbucket[b];
  const long base = (long)b * NN + (long)cb * CHUNK;
  const float* sc = smap + base;
  const unsigned sb = lds_off(sbuf);
  for (int q = 0; q < (int)(CHUNK * 4 / (TPB * 16)); ++q) {  // 4 issues/thread
    unsigned off = (unsigned)(t * 16 + q * TPB * 16);
    async_copy16(sb + off, sc, off);
  }
  wait_async0();
  __syncthreads();
  unsigned* fh = fine + (long)b * FINE_BINS;
  for (int k = 0; k < EPT; ++k) {
    unsigned key = __float_as_uint(sbuf[k * TPB + t]);
    if ((key >> 16) == tb) atomicAdd(&fh[key & 0xFFFFu], 1u);
  }
}

// Pass B2b: exact 32-bit threshold key + number of ties R to take.
__global__ void __launch_bounds__(TPB)
k_fine_select(const unsigned* __restrict__ fine, const int* __restrict__ Kp,
              const unsigned* __restrict__ Tbucket, const unsigned* __restrict__ G0,
              unsigned* __restrict__ Tkey, int* __restrict__ Rarr) {
  __shared__ unsigned csum[TPB];
  const int b = blockIdx.x, t = threadIdx.x;
  const unsigned* h = fine + (long)b * FINE_BINS;
  const int per = FINE_BINS / TPB;  // 256
  unsigned s = 0;
  for (int i = 0; i < per; ++i) s += h[t * per + i];
  csum[t] = s;
  __syncthreads();
  if (t == 0) {
    const unsigned Kr = (unsigned)Kp[0] - G0[b];  // >= 1 by construction
    unsigned acc = 0; int tc = 0;
    for (int c = TPB - 1; c >= 0; --c) {
      if (acc + csum[c] >= Kr) { tc = c; break; }
      acc += csum[c];
    }
    unsigned binSel = 0;
    for (int bin = (tc + 1) * per - 1; bin >= tc * per; --bin) {
      unsigned v = h[bin];
      if (acc + v >= Kr) { binSel = (unsigned)bin; break; }
      acc += v;
    }
    Tkey[b] = (Tbucket[b] << 16) | binSel;
    Rarr[b] = (int)(Kr - acc);  // ties to include, lowest indices first
  }
}

// Pass C: Sum(delta) over keys strictly above threshold; collect ties.
// Both streams (s_map keys + delta) staged via async global->LDS.
__global__ void __launch_bounds__(TPB)
k_select_sum(const float* __restrict__ smap, const float* __restrict__ delta,
             const unsigned* __restrict__ Tkey, unsigned* __restrict__ tieCnt,
             unsigned* __restrict__ tieIdx, float* __restrict__ tieDelta,
             float* __restrict__ pC) {
  __shared__ float sbuf[CHUNK];                 // 16 KB
  __shared__ float dbuf[CHUNK];                 // 16 KB
  __shared__ float lred[TPB / 32];
  const int t  = threadIdx.x;
  const int b  = blockIdx.x / BPB;
  const int cb = blockIdx.x % BPB;
  const unsigned T = Tkey[b];
  const long base = (long)b * NN + (long)cb * CHUNK;
  const float* sc = smap + base;
  const float* dc = delta + base;
  const unsigned sb = lds_off(sbuf);
  const unsigned db = lds_off(dbuf);
  for (int q = 0; q < (int)(CHUNK * 4 / (TPB * 16)); ++q) {  // 4+4 issues/thread
    unsigned off = (unsigned)(t * 16 + q * TPB * 16);
    async_copy16(sb + off, sc, off);
    async_copy16(db + off, dc, off);
  }
  wait_async0();
  __syncthreads();

  float acc0 = 0.f, acc1 = 0.f;
  for (int k = 0; k < EPT; ++k) {
    const int i = k * TPB + t;
    unsigned key = __float_as_uint(sbuf[i]);
    float dl = dbuf[i];
    float add = (key > T) ? dl : 0.f;
    if (k & 1) acc1 += add; else acc0 += add;
    if (key == T) {
      unsigned pos = atomicAdd(&tieCnt[b], 1u);
      if (pos < TIE_CAP) {
        tieIdx[(long)b * TIE_CAP + pos]   = (unsigned)(cb * CHUNK + i);
        tieDelta[(long)b * TIE_CAP + pos] = dl;
      }
    }
  }
  float tot = block_sum(acc0, acc1, lred);
  if (t == 0) pC[blockIdx.x] = tot;
}

// Final: deterministic fixed-order sums of partials, tie resolution, loss.
__global__ void __launch_bounds__(TPB)
k_final(const float* __restrict__ pA, const float* __restrict__ pC,
        const int* __restrict__ Rarr, const unsigned* __restrict__ tieCnt,
        const unsigned* __restrict__ tieIdx, const float* __restrict__ tieDelta,
        float* __restrict__ out) {
  __shared__ float red[TPB];
  const int t = threadIdx.x;
  float sa = 0.f, sc = 0.f;
  for (int i = t; i < NBLK; i += TPB) { sa += pA[i]; sc += pC[i]; }
  red[t] = sa; __syncthreads();
  for (int s = TPB / 2; s > 0; s >>= 1) { if (t < s) red[t] += red[t + s]; __syncthreads(); }
  float S1 = red[0]; __syncthreads();
  red[t] = sc; __syncthreads();
  for (int s = TPB / 2; s > 0; s >>= 1) { if (t < s) red[t] += red[t + s]; __syncthreads(); }
  if (t == 0) {
    float sel = red[0];
    for (int b = 0; b < BB; ++b) {
      int c = (int)tieCnt[b]; if (c > TIE_CAP) c = TIE_CAP;
      int R = Rarr[b];        if (R > c) R = c;
      unsigned prev = 0u;
      for (int r = 0; r < R; ++r) {   // take R ties with smallest indices
        unsigned best = 0xFFFFFFFFu; int bi = -1;
        for (int i = 0; i < c; ++i) {
          unsigned idx = tieIdx[b * TIE_CAP + i];
          if ((r == 0 || idx > prev) && idx < best) { best = idx; bi = i; }
        }
        if (bi < 0) break;
        sel += tieDelta[b * TIE_CAP + bi];
        prev = best;
      }
    }
    out[0] = -(S1 + sel);
  }
}

extern "C" void kernel_launch(void* const* d_in, const int* in_sizes, int n_in,
                              void* d_out, int out_size, void* d_ws, size_t ws_size,
                              hipStream_t stream) {
  (void)in_sizes; (void)n_in; (void)out_size; (void)ws_size;
  const float* ds   = (const float*)d_in[0];   // [B,N,2]
  const float* smap = (const float*)d_in[1];   // [B,1,H,W] -> [B,N]
  const float* U    = (const float*)d_in[2];   // [B,N,2]
  const int*   Kp   = (const int*)d_in[3];     // scalar K
  float* out = (float*)d_out;

  // workspace layout (~44.1 MB)
  float*    delta    = (float*)d_ws;                              // B*N f32
  unsigned* coarse   = (unsigned*)(delta + (size_t)BB * NN);      // B*16384 u32
  unsigned* fine     = coarse + (size_t)BB * COARSE_BINS;         // B*65536 u32
  float*    pA       = (float*)(fine + (size_t)BB * FINE_BINS);   // 2048 f32
  float*    pC       = pA + NBLK;                                 // 2048 f32
  unsigned* Tbucket  = (unsigned*)(pC + NBLK);
  unsigned* G0       = Tbucket + BB;
  unsigned* Tkey     = G0 + BB;
  int*      Rarr     = (int*)(Tkey + BB);
  unsigned* tieCnt   = (unsigned*)(Rarr + BB);
  unsigned* tieIdx   = tieCnt + BB;                               // B*256 u32
  float*    tieDelta = (float*)(tieIdx + (size_t)BB * TIE_CAP);   // B*256 f32

  const long zero_words = (long)BB * COARSE_BINS + (long)BB * FINE_BINS +
                          2L * NBLK + 5L * BB + 2L * BB * TIE_CAP;
  k_zero<<<(int)((zero_words + TPB - 1) / TPB), TPB, 0, stream>>>(coarse, zero_words);
  k_main<<<NBLK, TPB, 0, stream>>>(ds, smap, U, delta, coarse, pA);
  k_coarse_select<<<BB, TPB, 0, stream>>>(coarse, Kp, Tbucket, G0);
  k_fine_hist<<<NBLK, TPB, 0, stream>>>(smap, Tbucket, fine);
  k_fine_select<<<BB, TPB, 0, stream>>>(fine, Kp, Tbucket, G0, Tkey, Rarr);
  k_select_sum<<<NBLK, TPB, 0, stream>>>(smap, delta, Tkey, tieCnt, tieIdx, tieDelta, pC);
  k_final<<<1, TPB, 0, stream>>>(pA, pC, Rarr, tieCnt, tieIdx, tieDelta, out);
}